// TrainRoad2vecModel_75411035783384
// MI455X (gfx1250) — compile-verified
//
#include <hip/hip_runtime.h>
#include <math.h>

#define VOCAB 100000
#define EMB   128
#define EPS   1e-12f
#define ROWS  16               // items per wave tile
#define RSTR  (EMB + 4)        // padded LDS row stride in dwords (132 -> conflict free)
#define NWAVE 2                // waves per block
#define ITEMS_PER_BLOCK (ROWS * NWAVE)

typedef __attribute__((ext_vector_type(2))) float v2f;
typedef __attribute__((ext_vector_type(8))) float v8f;

// Low 32 bits of a generic LDS address are the LDS byte offset.
__device__ __forceinline__ uint32_t lds_offset(const void* p) {
    return (uint32_t)(uintptr_t)p;
}

// gfx1250 async copy global -> LDS, 16B per lane, tracked by ASYNCcnt (GV mode).
__device__ __forceinline__ void async_load_b128(uint32_t lds_off, const void* gptr) {
    asm volatile("global_load_async_to_lds_b128 %0, %1, off"
                 :: "v"(lds_off), "v"((uint64_t)(uintptr_t)gptr)
                 : "memory");
}

__device__ __forceinline__ void wait_async0() {
    asm volatile("s_wait_asynccnt 0" ::: "memory");
}

// ---------------------------------------------------------------------------
// Transpose W [EMB, VOCAB] -> Wt [VOCAB, EMB] with the encoder bias folded in:
//   Wt[v][k] = W[k][v] + b[k]
// Gathers of Wt rows are then 512B-contiguous pure copies.
// VOCAB = 3125*32, EMB = 4*32 -> exact tiling.
// ---------------------------------------------------------------------------
__global__ __launch_bounds__(1024)
void transpose_W_bias(const float* __restrict__ W, const float* __restrict__ b,
                      float* __restrict__ Wt) {
    __shared__ float tile[32][33];
    const int v0 = blockIdx.x * 32;
    const int k0 = blockIdx.y * 32;
    const int tx = threadIdx.x, ty = threadIdx.y;
    tile[ty][tx] = W[(size_t)(k0 + ty) * VOCAB + (v0 + tx)];        // coalesced read
    __syncthreads();
    Wt[(size_t)(v0 + ty) * EMB + (k0 + tx)] = tile[tx][ty] + b[k0 + tx]; // coalesced write
}

// ---------------------------------------------------------------------------
// Fused encode + normalize + heads. One wave handles 16 items via f32 WMMA:
//   c_xx = Ex*Ex^T  (diag -> ||ex||^2)
//   c_yy = Ey*Ey^T  (diag -> ||ey||^2)
//   c_p  = (Ex.*Ey) * [w1 w2 0...0]  (cols 0,1 -> head dots)
// ---------------------------------------------------------------------------
template <bool TRANSPOSED>
__global__ __launch_bounds__(32 * NWAVE)
void fused_road2vec(const int* __restrict__ x, const int* __restrict__ y,
                    const float* __restrict__ Wsrc,   // Wt(+bias) if TRANSPOSED, else W
                    const float* __restrict__ b,
                    const float* __restrict__ w1, const float* __restrict__ b1,
                    const float* __restrict__ w2, const float* __restrict__ b2,
                    float* __restrict__ out, int N) {
    __shared__ float sEx[NWAVE][ROWS * RSTR];
    __shared__ float sEy[NWAVE][ROWS * RSTR];
    __shared__ float sW[3 * EMB];          // w1 | w2 | zeros  (zero row kills cndmasks)
    __shared__ float sC[NWAVE][3][256];    // C-matrix bounce buffers

    const int tid  = threadIdx.x;
    const int wid  = tid >> 5;
    const int lane = tid & 31;
    const int tileBase = blockIdx.x * ITEMS_PER_BLOCK + wid * ROWS;

    // stage head weights + zero row
    for (int i = tid; i < EMB; i += 32 * NWAVE) {
        sW[i]           = w1[i];
        sW[EMB + i]     = w2[i];
        sW[2 * EMB + i] = 0.0f;
    }

    // ---- stage the two 16x128 embedding tiles into LDS ----
    if (TRANSPOSED) {
        // bias already folded into Wt: pure async copies, no VGPR round-trip
        #pragma unroll
        for (int r = 0; r < ROWS; ++r) {
            int item = tileBase + r; if (item >= N) item = N - 1;
            const int ix = x[item];
            const int iy = y[item];
            async_load_b128(lds_offset(&sEx[wid][r * RSTR + 4 * lane]),
                            Wsrc + (size_t)ix * EMB + 4 * lane);
            async_load_b128(lds_offset(&sEy[wid][r * RSTR + 4 * lane]),
                            Wsrc + (size_t)iy * EMB + 4 * lane);
        }
        wait_async0();
    } else {
        // fallback: strided column gather from W, bias added on the fly
        const float4 bv = *(const float4*)(b + 4 * lane);
        #pragma unroll 4
        for (int r = 0; r < ROWS; ++r) {
            int item = tileBase + r; if (item >= N) item = N - 1;
            const int ix = x[item];
            const int iy = y[item];
            const int c = 4 * lane;
            float4 vx, vy;
            vx.x = Wsrc[(size_t)(c + 0) * VOCAB + ix];
            vx.y = Wsrc[(size_t)(c + 1) * VOCAB + ix];
            vx.z = Wsrc[(size_t)(c + 2) * VOCAB + ix];
            vx.w = Wsrc[(size_t)(c + 3) * VOCAB + ix];
            vy.x = Wsrc[(size_t)(c + 0) * VOCAB + iy];
            vy.y = Wsrc[(size_t)(c + 1) * VOCAB + iy];
            vy.z = Wsrc[(size_t)(c + 2) * VOCAB + iy];
            vy.w = Wsrc[(size_t)(c + 3) * VOCAB + iy];
            vx.x += bv.x; vx.y += bv.y; vx.z += bv.z; vx.w += bv.w;
            vy.x += bv.x; vy.y += bv.y; vy.z += bv.z; vy.w += bv.w;
            *(float4*)(&sEx[wid][r * RSTR + 4 * lane]) = vx;
            *(float4*)(&sEy[wid][r * RSTR + 4 * lane]) = vy;
        }
    }
    __syncthreads();

    // ---- K-loop: 32 steps of V_WMMA_F32_16X16X4_F32 per chain ----
    // f32 16x4 A layout: lanes 0-15 hold M=0..15 with K = k0,k0+1 in the two
    // VGPRs; lanes 16-31 hold K = k0+2,k0+3. 4x16 B mirrors it, so the Gram
    // chains reuse the A register pair directly.
    v8f c_xx = {}, c_yy = {}, c_p = {};
    const int m    = lane & 15;
    const int khal = (lane & 16) ? 2 : 0;
    // per-lane B row: lane n==0 reads w1, n==1 reads w2, others read zeros
    const float* wrow = &sW[(m == 0) ? 0 : ((m == 1) ? EMB : 2 * EMB)];

    #pragma unroll 4
    for (int it = 0; it < EMB / 4; ++it) {
        const int kA = 4 * it + khal;
        const v2f ax = *(const v2f*)(&sEx[wid][m * RSTR + kA]);
        const v2f ay = *(const v2f*)(&sEy[wid][m * RSTR + kA]);

        c_xx = __builtin_amdgcn_wmma_f32_16x16x4_f32(false, ax, false, ax,
                                                     (short)0, c_xx, false, false);
        c_yy = __builtin_amdgcn_wmma_f32_16x16x4_f32(false, ay, false, ay,
                                                     (short)0, c_yy, false, false);

        const v2f q  = ax * ay;                       // (ex .* ey) chunk
        const v2f bw = *(const v2f*)(&wrow[kA]);      // single broadcast LDS load
        c_p = __builtin_amdgcn_wmma_f32_16x16x4_f32(false, q, false, bw,
                                                    (short)0, c_p, false, false);
    }

    // ---- spill C matrices to LDS to extract diagonals / columns ----
    // C layout: VGPR r, lanes 0-15 -> (M=r, N=lane); lanes 16-31 -> (M=r+8).
    const int Mofs = (lane & 16) ? 8 : 0;
    #pragma unroll
    for (int r = 0; r < 8; ++r) {
        const int idx = (r + Mofs) * 16 + m;
        sC[wid][0][idx] = c_xx[r];
        sC[wid][1][idx] = c_yy[r];
        sC[wid][2][idx] = c_p[r];
    }
    __syncthreads();

    if (lane < 16) {
        const int item = tileBase + lane;
        if (item < N) {
            const float sxx = sC[wid][0][lane * 17];      // diag
            const float syy = sC[wid][1][lane * 17];      // diag
            const float p1  = sC[wid][2][lane * 16 + 0];  // col 0
            const float p2  = sC[wid][2][lane * 16 + 1];  // col 1
            const float nx  = fmaxf(sqrtf(sxx), EPS);
            const float ny  = fmaxf(sqrtf(syy), EPS);
            const float inv = 1.0f / (nx * ny);
            const float z1  = p1 * inv + b1[0];
            const float z2  = p2 * inv + b2[0];
            out[item]     = 1.0f / (1.0f + __expf(-z1));
            out[N + item] = 1.0f / (1.0f + __expf(-z2));
        }
    }
}

// ---------------------------------------------------------------------------
extern "C" void kernel_launch(void* const* d_in, const int* in_sizes, int n_in,
                              void* d_out, int out_size, void* d_ws, size_t ws_size,
                              hipStream_t stream) {
    const int*   x  = (const int*)d_in[0];
    const int*   y  = (const int*)d_in[1];
    const float* W  = (const float*)d_in[2];
    const float* b  = (const float*)d_in[3];
    const float* w1 = (const float*)d_in[4];
    const float* b1 = (const float*)d_in[5];
    const float* w2 = (const float*)d_in[6];
    const float* b2 = (const float*)d_in[7];
    float*       out = (float*)d_out;
    const int    N   = in_sizes[0];

    const size_t needWt = (size_t)VOCAB * EMB * sizeof(float);
    const int nBlocks = (N + ITEMS_PER_BLOCK - 1) / ITEMS_PER_BLOCK;

    if (ws_size >= needWt) {
        float* Wt = (float*)d_ws;
        transpose_W_bias<<<dim3(VOCAB / 32, EMB / 32), dim3(32, 32), 0, stream>>>(W, b, Wt);
        fused_road2vec<true><<<nBlocks, 32 * NWAVE, 0, stream>>>(
            x, y, Wt, b, w1, b1, w2, b2, out, N);
    } else {
        fused_road2vec<false><<<nBlocks, 32 * NWAVE, 0, stream>>>(
            x, y, W, b, w1, b1, w2, b2, out, N);
    }
}